// GeToInformedNeighborSampler_45612552683931
// MI455X (gfx1250) — compile-verified
//
#include <hip/hip_runtime.h>
#include <math.h>

typedef __attribute__((ext_vector_type(2))) float v2f;
typedef __attribute__((ext_vector_type(8))) float v8f;
typedef unsigned int       u32;
typedef unsigned long long u64;

// ---------------------------------------------------------------------------
// threefry2x32-20, JAX flavor. key = (0, 42) from jax.random.key(42).
// ---------------------------------------------------------------------------
__device__ __forceinline__ void tf_round(u32& x0, u32& x1, int r) {
  x0 += x1;
  x1 = (x1 << r) | (x1 >> (32 - r));
  x1 ^= x0;
}

__device__ __forceinline__ u32 threefry_out(u32 c0, u32 c1, int lane) {
  const u32 ks0 = 0u, ks1 = 42u, ks2 = 0x1BD11BDAu ^ 0u ^ 42u;
  u32 x0 = c0 + ks0, x1 = c1 + ks1;
  tf_round(x0, x1, 13); tf_round(x0, x1, 15); tf_round(x0, x1, 26); tf_round(x0, x1, 6);
  x0 += ks1; x1 += ks2 + 1u;
  tf_round(x0, x1, 17); tf_round(x0, x1, 29); tf_round(x0, x1, 16); tf_round(x0, x1, 24);
  x0 += ks2; x1 += ks0 + 2u;
  tf_round(x0, x1, 13); tf_round(x0, x1, 15); tf_round(x0, x1, 26); tf_round(x0, x1, 6);
  x0 += ks0; x1 += ks1 + 3u;
  tf_round(x0, x1, 17); tf_round(x0, x1, 29); tf_round(x0, x1, 16); tf_round(x0, x1, 24);
  x0 += ks1; x1 += ks2 + 4u;
  tf_round(x0, x1, 13); tf_round(x0, x1, 15); tf_round(x0, x1, 26); tf_round(x0, x1, 6);
  x0 += ks2; x1 += ks0 + 5u;
  return lane ? x1 : x0;
}

// bits for flat element f of the (64, B*64) gumbel array (JAX half-split pairing)
__device__ __forceinline__ u32 random_bits(u64 f, u64 half) {
  u32 c0, c1; int lane;
  if (f < half) { c0 = (u32)f;          c1 = (u32)(f + half); lane = 0; }
  else          { c0 = (u32)(f - half); c1 = (u32)f;          lane = 1; }
  return threefry_out(c0, c1, lane);
}

__device__ __forceinline__ float gumbel_from_bits(u32 bits) {
  // JAX uniform(minval=tiny, maxval=1): bits>>9 | 1.0f exponent, -1, scale, max
  float u = __uint_as_float((bits >> 9) | 0x3F800000u) - 1.0f;
  const float tiny = 1.1754943508222875e-38f;
  u = u * (1.0f - tiny) + tiny;
  u = fmaxf(u, tiny);
  return -logf(-logf(u));
}

// monotone float -> orderable u32
__device__ __forceinline__ u32 f2ord(float f) {
  u32 b = __float_as_uint(f);
  return (b & 0x80000000u) ? ~b : (b | 0x80000000u);
}

__device__ __forceinline__ u64 shfl_down_u64(u64 v, int d) {
  u32 lo = (u32)v, hi = (u32)(v >> 32);
  lo = __shfl_down(lo, d, 32);
  hi = __shfl_down(hi, d, 32);
  return ((u64)hi << 32) | lo;
}

// ---------------------------------------------------------------------------
// Kernel 0: re-init workspace argmax slots every call (ws is poisoned once).
// ---------------------------------------------------------------------------
__global__ void k_init(u64* packed) {
  if (threadIdx.x < 32) packed[threadIdx.x] = 0ull;
}

// ---------------------------------------------------------------------------
// Kernel 1: Gumbel-argmax per sample s = blockIdx.y over j in [0, M).
// key = (ordered(logit+gumbel) << 32) | ~j  -> atomicMax gives JAX tie-break
// (first occurrence of the max).
// ---------------------------------------------------------------------------
__global__ void k_argmax(const int* __restrict__ ids,
                         const int* __restrict__ geto,
                         const float* __restrict__ probs,
                         u64* __restrict__ packed,
                         u32 M, int deg, u64 half) {
  const int s = blockIdx.y;
  u64 best = 0ull;
  const u32 stride = gridDim.x * blockDim.x;
  for (u32 j = blockIdx.x * blockDim.x + threadIdx.x; j < M; j += stride) {
    u32 row = j / (u32)deg;
    u32 d   = j - row * (u32)deg;
    int nb  = geto[(size_t)(u32)ids[row] * (u32)deg + d];
    float logit = logf(probs[nb]);
    u64 f = (u64)s * (u64)M + (u64)j;
    float val = logit + gumbel_from_bits(random_bits(f, half));
    u64 key = ((u64)f2ord(val) << 32) | (u32)(~j);
    best = key > best ? key : best;
  }
  // wave32 reduce
  for (int off = 16; off > 0; off >>= 1) {
    u64 o = shfl_down_u64(best, off);
    best = o > best ? o : best;
  }
  __shared__ u64 wred[8];
  const int lane = threadIdx.x & 31, wave = threadIdx.x >> 5;
  if (lane == 0) wred[wave] = best;
  __syncthreads();
  if (wave == 0) {
    int nw = (blockDim.x + 31) >> 5;
    u64 v = (lane < nw) ? wred[lane] : 0ull;
    for (int off = 4; off > 0; off >>= 1) {
      u64 o = shfl_down_u64(v, off);
      v = o > v ? o : v;
    }
    if (lane == 0) atomicMax(&packed[s], v);
  }
}

// ---------------------------------------------------------------------------
// Kernel 2: decode winners, apply 'clip' mode (-> [0, deg-1]); sentinel for
// padding columns so the selection matrix column is all-zero.
// ---------------------------------------------------------------------------
__global__ void k_decode(const u64* __restrict__ packed, int* __restrict__ cidx,
                         int ns, int deg) {
  int t = threadIdx.x;
  if (t < 32) {
    if (t < ns) {
      u32 j = ~((u32)packed[t]);              // winning flat index
      u32 c = j < (u32)deg ? j : (u32)(deg - 1);
      cidx[t] = (int)c;
    } else {
      cidx[t] = 0x7FFFFFFF;                   // matches no k in [0,64)
    }
  }
}

// ---------------------------------------------------------------------------
// Kernel 3: exact gather-as-matmul. Each wave owns a 16-row strip:
//   out[16 x 32] = A[16 x 64] x S[64 x 32],   S[k][n] = (c[n] == k)
// done as 16 chained V_WMMA_F32_16X16X4_F32 per 16-col tile, for both the
// adj and geto tables (64 WMMAs / wave). f32 WMMA is exact: ids < 2^24 and
// exactly one nonzero per dot product. The 32-entry c table is staged into
// LDS with the CDNA5 async global->LDS path.
// ---------------------------------------------------------------------------
__device__ __forceinline__ v8f wmma4(v2f a, v2f b, v8f c) {
  return __builtin_amdgcn_wmma_f32_16x16x4_f32(false, a, false, b, (short)0, c,
                                               false, false);
}

__global__ void k_gather_wmma(const int* __restrict__ ids,
                              const int* __restrict__ adj,
                              const int* __restrict__ geto,
                              const int* __restrict__ cidx,
                              float* __restrict__ outA,
                              float* __restrict__ outG,
                              int B, int deg, int ns) {
  __shared__ int sC[32];
  if (threadIdx.x < 32) {
    // async copy of the clipped-index table into LDS (ASYNCcnt-tracked)
    unsigned ldsOff = (unsigned)(size_t)(void*)&sC[threadIdx.x];
    const int* gp = cidx + threadIdx.x;
    asm volatile("global_load_async_to_lds_b32 %0, %1, off"
                 :: "v"(ldsOff), "v"(gp) : "memory");
    asm volatile("s_wait_asynccnt 0" ::: "memory");
  }
  __syncthreads();

  const int lane = threadIdx.x & 31;
  const int m    = lane & 15;      // M index within tile
  const int hi   = lane >> 4;      // lane half selects K sub-pair
  const int strip = blockIdx.x * (blockDim.x >> 5) + (threadIdx.x >> 5);
  const int rowBase = strip * 16;
  if (rowBase >= B) return;        // uniform per wave

  int row = rowBase + m;
  if (row >= B) row = B - 1;       // clamp loads; stores guarded (EXEC stays full)
  const size_t rbase = (size_t)(u32)ids[row] * (u32)deg;
  const int* __restrict__ arow = adj  + rbase;
  const int* __restrict__ grow = geto + rbase;

  const int c0 = sC[m];            // selection targets for col tiles 0 and 1
  const int c1 = sC[16 + m];

  v8f accA0 = {}, accA1 = {}, accG0 = {}, accG1 = {};
  for (int k0 = 0; k0 < deg; k0 += 4) {
    const int kk = k0 + hi * 2;    // this lane's K pair (A layout: K = 2*hi + r)
    v2f a, g, b0, b1;
    a[0] = (float)arow[kk];  a[1] = (float)arow[kk + 1];
    g[0] = (float)grow[kk];  g[1] = (float)grow[kk + 1];
    b0[0] = (c0 == kk) ? 1.0f : 0.0f;  b0[1] = (c0 == kk + 1) ? 1.0f : 0.0f;
    b1[0] = (c1 == kk) ? 1.0f : 0.0f;  b1[1] = (c1 == kk + 1) ? 1.0f : 0.0f;
    accA0 = wmma4(a, b0, accA0);
    accA1 = wmma4(a, b1, accA1);
    accG0 = wmma4(g, b0, accG0);
    accG1 = wmma4(g, b1, accG1);
  }

  // C layout: reg r, lane half hi -> M = r + 8*hi, N = lane&15
  for (int r = 0; r < 8; ++r) {
    const int orow = rowBase + r + 8 * hi;
    if (orow < B) {
      const size_t ob = (size_t)orow * (u32)ns;
      if (m < ns)      { outA[ob + m]      = accA0[r]; outG[ob + m]      = accG0[r]; }
      if (16 + m < ns) { outA[ob + 16 + m] = accA1[r]; outG[ob + 16 + m] = accG1[r]; }
    }
  }
}

// ---------------------------------------------------------------------------
extern "C" void kernel_launch(void* const* d_in, const int* in_sizes, int n_in,
                              void* d_out, int out_size, void* d_ws, size_t ws_size,
                              hipStream_t stream) {
  const int*   ids   = (const int*)d_in[0];
  const int*   adj   = (const int*)d_in[2];
  const int*   geto  = (const int*)d_in[3];
  const float* probs = (const float*)d_in[4];

  const int B   = in_sizes[0];
  const int deg = in_sizes[2] / B;              // 64
  const int ns  = out_size / (2 * B);           // 25

  u64* packed = (u64*)d_ws;
  int* cidx   = (int*)((char*)d_ws + 32 * sizeof(u64));

  const u32 M    = (u32)B * (u32)deg;           // 6.4M logits
  const u64 half = ((u64)deg * (u64)M) >> 1;    // threefry half-split point

  k_init<<<1, 32, 0, stream>>>(packed);

  dim3 g1(512, (unsigned)ns);
  k_argmax<<<g1, 256, 0, stream>>>(ids, geto, probs, packed, M, deg, half);

  k_decode<<<1, 32, 0, stream>>>(packed, cidx, ns, deg);

  float* outA = (float*)d_out;
  float* outG = outA + (size_t)B * (u32)ns;
  const int strips = (B + 15) / 16;
  const int wavesPerBlock = 8;
  const int blocks = (strips + wavesPerBlock - 1) / wavesPerBlock;
  k_gather_wmma<<<blocks, 32 * wavesPerBlock, 0, stream>>>(
      ids, adj, geto, cidx, outA, outG, B, deg, ns);
}